// FNOBlockNd_NBF_65515431133932
// MI455X (gfx1250) — compile-verified
//
#include <hip/hip_runtime.h>
#include <math.h>
#include <stdint.h>

// ---------------------------------------------------------------------------
// FNO block for MI455X (gfx1250, wave32, WMMA f32 16x16x4).
// Partial-DFT factorization: all heavy stages are fp32 WMMA GEMMs.
// LDS panels filled with gfx1250 async global->LDS copies (ASYNCcnt).
// ---------------------------------------------------------------------------

typedef __attribute__((ext_vector_type(2))) float v2f;
typedef __attribute__((ext_vector_type(8))) float v8f;

// workspace layout (float offsets)
constexpr size_t OFF_EH   = 0;                      // [64][256] fwd H basis (cos rows 0..31, -sin rows 32..63)
constexpr size_t OFF_CW   = OFF_EH  + 64 * 256;     // [32][256] cos(2pi l w/256)/256
constexpr size_t OFF_SW   = OFF_CW  + 32 * 256;     // [32][256] sin(...)/256
constexpr size_t OFF_CHT  = OFF_SW  + 32 * 256;     // [32][256] inv basis cos, [k][h]
constexpr size_t OFF_SHT  = OFF_CHT + 32 * 256;     // [32][256] inv basis sin, [k][h]
constexpr size_t OFF_DD   = OFF_SHT + 32 * 256;     // [64][256] stacked c_l*cos/256 ; -c_l*sin/256
constexpr size_t OFF_G1P  = OFF_DD  + 64 * 256;     // [16][64] 1+gamma
constexpr size_t OFF_BETA = OFF_G1P + 1024;         // [16][64] beta
constexpr size_t OFF_XF   = OFF_BETA + 1024;        // [2][1024 modes][64 i][16 b]
constexpr size_t OFF_WT   = OFF_XF  + 2u * 1024 * 64 * 16;   // [2][1024 modes][64 i][64 o]
constexpr size_t OFF_YF   = OFF_WT  + 2u * 1024 * 64 * 64;   // [2][16 b][64 o][1024 mode(k*32+l)]
constexpr size_t OFF_T12  = OFF_YF  + 2u * 16 * 64 * 1024;   // T1:[1024 bi][64][256]  reused as T2:[1024 bo][2][256][32]

__device__ __forceinline__ v8f wmma4(v2f a, v2f b, v8f c) {
  // D = A(16x4,f32) * B(4x16,f32) + C(16x16,f32)   -> v_wmma_f32_16x16x4_f32
  return __builtin_amdgcn_wmma_f32_16x16x4_f32(false, a, false, b, (short)0, c,
                                               false, false);
}

// gfx1250 async global->LDS copy, 16 bytes per enabled lane (ASYNCcnt).
// VDST = LDS byte address VGPR, VADDR = 64-bit global address (GV mode).
__device__ __forceinline__ void async_copy16(void* lds_dst, const float* gsrc) {
  uint32_t loff = (uint32_t)(uintptr_t)lds_dst;
  asm volatile("global_load_async_to_lds_b128 %0, %1, off"
               :: "v"(loff), "v"(gsrc)
               : "memory");
}
__device__ __forceinline__ void wait_async0() {
  asm volatile("s_wait_asynccnt 0x0" ::: "memory");
}

// fragment loaders.  A-frag and B-frag share the same lane mapping:
//   lanes 0-15 -> M/N = mn0+lane, VGPR0 holds K=k(+2 for hi half), VGPR1 K=k+1(+2)
__device__ __forceinline__ v2f ld_mnmaj(const float* p, int ld, int mn0, int k) {
  int lane = threadIdx.x & 31;
  int r = mn0 + (lane & 15);
  int kk = k + ((lane >> 4) << 1);
  v2f v; v.x = p[r * ld + kk]; v.y = p[r * ld + kk + 1]; return v;
}
__device__ __forceinline__ v2f ld_kmaj(const float* p, int ld, int k, int mn0) {
  int lane = threadIdx.x & 31;
  int c = mn0 + (lane & 15);
  int kk = k + ((lane >> 4) << 1);
  v2f v; v.x = p[kk * ld + c]; v.y = p[(kk + 1) * ld + c]; return v;
}
__device__ __forceinline__ v2f neg2(v2f v) { v2f r; r.x = -v.x; r.y = -v.y; return r; }

__device__ __forceinline__ float gelu_tanh(float v) {
  const float c0 = 0.7978845608028654f; // sqrt(2/pi)
  float u = c0 * (v + 0.044715f * v * v * v);
  return 0.5f * v * (1.0f + tanhf(u));
}

// ---------------------------------------------------------------------------
// K0: twiddle tables (angles reduced mod 256 in integers -> exact phases)
// ---------------------------------------------------------------------------
__global__ void __launch_bounds__(256) k_tables(float* ws) {
  float* EH  = ws + OFF_EH;
  float* CW  = ws + OFF_CW;
  float* SW  = ws + OFF_SW;
  float* CHT = ws + OFF_CHT;
  float* SHT = ws + OFF_SHT;
  float* DD  = ws + OFF_DD;
  const float W2PI = 6.2831853071795864769f / 256.0f;
  const float INV = 1.0f / 256.0f;
  for (int idx = threadIdx.x; idx < 32 * 256; idx += 256) {
    int k = idx >> 8, h = idx & 255;
    int ph = (k * h) & 255;
    float c = cosf(W2PI * (float)ph);
    float s = sinf(W2PI * (float)ph);
    EH[k * 256 + h]        = c;      // Re e^{-i t}
    EH[(32 + k) * 256 + h] = -s;     // Im e^{-i t}
    CW[idx] = c * INV;
    SW[idx] = s * INV;
    CHT[idx] = c;                    // Re e^{+i t}
    SHT[idx] = s;                    // Im e^{+i t}
    float cl = (k == 0) ? 1.0f : 2.0f;   // Hermitian fold weight for irfft
    DD[k * 256 + h]        = cl * c * INV;
    DD[(32 + k) * 256 + h] = -cl * s * INV;
  }
}

// ---------------------------------------------------------------------------
// K1: FiLM  (tiny: 16x64 and 16x128 GEMVs, one workgroup)
// ---------------------------------------------------------------------------
__global__ void __launch_bounds__(256) k_film(const float* t_emb, const float* w1,
                                              const float* b1, const float* w2,
                                              const float* b2, float* ws) {
  __shared__ float hbuf[16 * 64];
  float* g1p  = ws + OFF_G1P;
  float* beta = ws + OFF_BETA;
  for (int idx = threadIdx.x; idx < 16 * 64; idx += 256) {
    int b = idx >> 6, c = idx & 63;
    float acc = b1[c];
    for (int i = 0; i < 64; ++i) acc += t_emb[b * 64 + i] * w1[i * 64 + c];
    hbuf[idx] = acc / (1.0f + __expf(-acc));   // silu
  }
  __syncthreads();
  for (int idx = threadIdx.x; idx < 16 * 128; idx += 256) {
    int b = idx >> 7, j = idx & 127;
    float acc = b2[j];
    for (int c = 0; c < 64; ++c) acc += hbuf[b * 64 + c] * w2[c * 128 + j];
    if (j < 64) g1p[b * 64 + j] = 1.0f + acc;
    else        beta[b * 64 + (j - 64)] = acc;
  }
}

// ---------------------------------------------------------------------------
// K2: transpose spectral weights -> Wt[part][mode][i][o] (coalesced for k_mix)
// grid: 2 * 64 * 16 blocks (part, i, mode-tile of 64)
// ---------------------------------------------------------------------------
__global__ void __launch_bounds__(256) k_wt(const float* wr, const float* wi, float* ws) {
  __shared__ float tile[64][65];
  float* wt = ws + OFF_WT;
  int blk = blockIdx.x;
  int mt = blk & 15;
  int i  = (blk >> 4) & 63;
  int p  = blk >> 10;
  const float* src = p ? wi : wr;
  int m0 = mt * 64;
  for (int idx = threadIdx.x; idx < 64 * 64; idx += 256) {
    int o = idx >> 6, mm = idx & 63;
    tile[o][mm] = src[((size_t)i * 64 + o) * 1024 + m0 + mm];
  }
  __syncthreads();
  for (int idx = threadIdx.x; idx < 64 * 64; idx += 256) {
    int mm = idx >> 6, o = idx & 63;
    wt[(((size_t)p * 1024 + m0 + mm) * 64 + i) * 64 + o] = tile[o][mm];
  }
}

// ---------------------------------------------------------------------------
// K3: forward H-DFT.  T1[bi](64x256) = EH(64x256) @ x[bi](256x256)
// grid (1024, 4): image x 64-wide w strip.  8 waves, 2 output tiles each.
// Panels streamed in with async global->LDS B128 copies.
// ---------------------------------------------------------------------------
__global__ void __launch_bounds__(256) k_fwdH(const float* x, float* ws) {
  __shared__ float eh[64][68];   // [M][K] padded (row stride 272B = 17*16B)
  __shared__ float xs[64][64];   // [K][N]
  const float* EH = ws + OFF_EH;
  float* T1 = ws + OFF_T12;
  int bi = blockIdx.x;
  int w0 = blockIdx.y * 64;
  int t = threadIdx.x, wid = t >> 5;
  int m = wid & 3;
  int n0 = (wid >> 2) << 1;
  const float* img = x + (size_t)bi * 65536;
  v8f acc0 = {}, acc1 = {};
  for (int k0 = 0; k0 < 256; k0 += 64) {
    __syncthreads();                       // prior panel fully consumed
#pragma unroll
    for (int idx = t; idx < 1024; idx += 256) {   // 1024 x 16B per panel
      int r = idx >> 4, c = (idx & 15) << 2;
      async_copy16(&eh[r][c], &EH[r * 256 + k0 + c]);
      async_copy16(&xs[r][c], &img[(k0 + r) * 256 + w0 + c]);
    }
    wait_async0();
    __syncthreads();
#pragma unroll
    for (int k = 0; k < 64; k += 4) {
      v2f a  = ld_mnmaj(&eh[0][0], 68, m * 16, k);
      v2f b0 = ld_kmaj(&xs[0][0], 64, k, n0 * 16);
      v2f b1 = ld_kmaj(&xs[0][0], 64, k, n0 * 16 + 16);
      acc0 = wmma4(a, b0, acc0);
      acc1 = wmma4(a, b1, acc1);
    }
  }
  float* out = T1 + (size_t)bi * 16384 + w0;   // [64][256] row-major
  int lane = t & 31, nn = lane & 15, mh = (lane >> 4) * 8;
#pragma unroll
  for (int r = 0; r < 8; ++r) {
    out[(m * 16 + mh + r) * 256 + n0 * 16 + nn]      = acc0[r];
    out[(m * 16 + mh + r) * 256 + n0 * 16 + 16 + nn] = acc1[r];
  }
}

// ---------------------------------------------------------------------------
// K4: forward W-DFT.  Xf(32x32 cplx) = T1 * conj basis.  grid 1024 images.
//   Xfr = T1r@CW^T + T1i@SW^T ; Xfi = T1i@CW^T - T1r@SW^T   (1/256 in tables)
// ---------------------------------------------------------------------------
__global__ void __launch_bounds__(256) k_fwdW(float* ws) {
  __shared__ float t1[64][68];   // [M][K] padded (rows 0-31 real, 32-63 imag)
  __shared__ float cw[32][68];   // [N][K] padded
  __shared__ float sw[32][68];
  const float* T1 = ws + OFF_T12;
  const float* CW = ws + OFF_CW;
  const float* SW = ws + OFF_SW;
  float* Xf = ws + OFF_XF;
  int bi = blockIdx.x;
  int t = threadIdx.x, wid = t >> 5, lane = t & 31;
  int p = wid >> 2;          // 0 -> Xfr, 1 -> Xfi
  int m = (wid >> 1) & 1;    // k-row tile
  int n = wid & 1;           // l-col tile
  v8f acc = {};
  for (int k0 = 0; k0 < 256; k0 += 64) {
    __syncthreads();
#pragma unroll
    for (int idx = t; idx < 1024; idx += 256) {
      int r = idx >> 4, c = (idx & 15) << 2;
      async_copy16(&t1[r][c], &T1[(size_t)bi * 16384 + r * 256 + k0 + c]);
    }
    {
      int idx = t & 511;                       // 512 x 16B per table panel
      int r = idx >> 4, c = (idx & 15) << 2;
      if (t < 512) {
        async_copy16(&cw[r][c], &CW[r * 256 + k0 + c]);
      } else {
        async_copy16(&sw[r][c], &SW[r * 256 + k0 + c]);
      }
    }
    wait_async0();
    __syncthreads();
#pragma unroll
    for (int k = 0; k < 64; k += 4) {
      v2f ar = ld_mnmaj(&t1[0][0], 68, m * 16, k);
      v2f ai = ld_mnmaj(&t1[0][0], 68, 32 + m * 16, k);
      v2f bc = ld_mnmaj(&cw[0][0], 68, n * 16, k);   // B stored [N][K]
      v2f bs = ld_mnmaj(&sw[0][0], 68, n * 16, k);
      if (p == 0) {                 // wave-uniform branch
        acc = wmma4(ar, bc, acc);
        acc = wmma4(ai, bs, acc);
      } else {
        acc = wmma4(ai, bc, acc);
        acc = wmma4(ar, neg2(bs), acc);
      }
    }
  }
  // Xf[p][mode][i][b], mode = krow*32 + lcol
  int nn = n * 16 + (lane & 15);
  int mh = (lane >> 4) * 8;
  int b = bi >> 6, ci = bi & 63;
#pragma unroll
  for (int r = 0; r < 8; ++r) {
    int mode = (m * 16 + mh + r) * 32 + nn;
    Xf[(((size_t)p * 1024 + mode) * 64 + ci) * 16 + b] = acc[r];
  }
}

// ---------------------------------------------------------------------------
// K5: per-mode complex channel mix.  Y(16x64) = X(16x64) * W(64x64), complex.
// grid 1024 modes. 8 waves: {Yr,Yi} x 4 n-tiles.
// ---------------------------------------------------------------------------
__global__ void __launch_bounds__(256) k_mix(float* ws) {
  const float* Xf = ws + OFF_XF;
  const float* Wt = ws + OFF_WT;
  float* Yf = ws + OFF_YF;
  int mode = blockIdx.x;
  int t = threadIdx.x, wid = t >> 5, lane = t & 31;
  int p = wid >> 2, n = wid & 3;
  const float* xr = Xf + (size_t)mode * 1024;
  const float* xi = Xf + (size_t)(1024 + mode) * 1024;
  const float* wr = Wt + (size_t)mode * 4096;
  const float* wi = Wt + (size_t)(1024 + mode) * 4096;
  v8f acc = {};
#pragma unroll
  for (int k = 0; k < 64; k += 4) {
    v2f ar = ld_kmaj(xr, 16, k, 0);          // [i][b] storage = K-major
    v2f ai = ld_kmaj(xi, 16, k, 0);
    v2f br = ld_kmaj(wr, 64, k, n * 16);     // [i][o] storage = K-major
    v2f bi = ld_kmaj(wi, 64, k, n * 16);
    if (p == 0) {                            // Yr = Xr*Wr - Xi*Wi
      acc = wmma4(ar, br, acc);
      acc = wmma4(ai, neg2(bi), acc);
    } else {                                 // Yi = Xr*Wi + Xi*Wr
      acc = wmma4(ar, bi, acc);
      acc = wmma4(ai, br, acc);
    }
  }
  int nn = n * 16 + (lane & 15);
  int mh = (lane >> 4) * 8;
#pragma unroll
  for (int r = 0; r < 8; ++r) {
    int b = mh + r;
    Yf[(((size_t)p * 16 + b) * 64 + nn) * 1024 + mode] = acc[r];
  }
}

// ---------------------------------------------------------------------------
// K6: inverse H-DFT.  T2(256x32 cplx) = e^{+i}H(256x32) @ Y(32x32 cplx)
// grid 1024 (b*64+o). 8 waves x 8 accumulators (2m x 2n x 2part).
// ---------------------------------------------------------------------------
__global__ void __launch_bounds__(256) k_invH(float* ws) {
  __shared__ float yr[32][32], yi[32][32];
  const float* Yf  = ws + OFF_YF;
  const float* CHT = ws + OFF_CHT;   // [k][h] -> K-major for A frags
  const float* SHT = ws + OFF_SHT;
  float* T2 = ws + OFF_T12;
  int bo = blockIdx.x;
  int t = threadIdx.x, wid = t >> 5, lane = t & 31;
  {
    int r = t >> 3, c = (t & 7) << 2;        // 256 threads x 16B covers 32x32
    async_copy16(&yr[r][c], &Yf[(size_t)bo * 1024 + r * 32 + c]);
    async_copy16(&yi[r][c], &Yf[(size_t)(1024 * 1024) + (size_t)bo * 1024 + r * 32 + c]);
  }
  wait_async0();
  __syncthreads();
  v8f acc[8] = {};
  int h0 = wid * 32;   // this wave's 32 output rows
#pragma unroll
  for (int k = 0; k < 32; k += 4) {
    v2f ac0 = ld_kmaj(CHT, 256, k, h0);
    v2f ac1 = ld_kmaj(CHT, 256, k, h0 + 16);
    v2f as0 = ld_kmaj(SHT, 256, k, h0);
    v2f as1 = ld_kmaj(SHT, 256, k, h0 + 16);
    v2f br0 = ld_kmaj(&yr[0][0], 32, k, 0);
    v2f br1 = ld_kmaj(&yr[0][0], 32, k, 16);
    v2f bi0 = ld_kmaj(&yi[0][0], 32, k, 0);
    v2f bi1 = ld_kmaj(&yi[0][0], 32, k, 16);
    v2f bi0n = neg2(bi0), bi1n = neg2(bi1);
    // T2r = Ch@Yr - Sh@Yi
    acc[0] = wmma4(ac0, br0, acc[0]); acc[0] = wmma4(as0, bi0n, acc[0]);
    acc[1] = wmma4(ac0, br1, acc[1]); acc[1] = wmma4(as0, bi1n, acc[1]);
    acc[2] = wmma4(ac1, br0, acc[2]); acc[2] = wmma4(as1, bi0n, acc[2]);
    acc[3] = wmma4(ac1, br1, acc[3]); acc[3] = wmma4(as1, bi1n, acc[3]);
    // T2i = Sh@Yr + Ch@Yi
    acc[4] = wmma4(as0, br0, acc[4]); acc[4] = wmma4(ac0, bi0, acc[4]);
    acc[5] = wmma4(as0, br1, acc[5]); acc[5] = wmma4(ac0, bi1, acc[5]);
    acc[6] = wmma4(as1, br0, acc[6]); acc[6] = wmma4(ac1, bi0, acc[6]);
    acc[7] = wmma4(as1, br1, acc[7]); acc[7] = wmma4(ac1, bi1, acc[7]);
  }
  // T2[bo][part][256][32]
  size_t base = (size_t)bo * 16384;
  int nloc = lane & 15, mh = (lane >> 4) * 8;
#pragma unroll
  for (int a = 0; a < 8; ++a) {
    int p  = a >> 2;
    int mi = (a >> 1) & 1;
    int ni = a & 1;
#pragma unroll
    for (int r = 0; r < 8; ++r) {
      int row = h0 + mi * 16 + mh + r;
      int col = ni * 16 + nloc;
      T2[base + (size_t)p * 8192 + row * 32 + col] = acc[a][r];
    }
  }
}

// ---------------------------------------------------------------------------
// K7: fused final stage per (b, h, 64-wide w tile):
//   acc  = bypass_w(64x64) @ x[b,:,h,wtile]           (bypass GEMM)
//        + [(1+g)*T2r | (1+g)*T2i](64x64) @ DD(64x64) (spectral reconstr + FiLM gain)
//   out  = gelu(acc + beta + bypass_b)
// ---------------------------------------------------------------------------
__global__ void __launch_bounds__(256) k_final(const float* x, const float* bw,
                                               const float* bb, const float* ws,
                                               float* out) {
  __shared__ float xs[64][64];   // [i][w]  K-major
  __shared__ float a2[64][68];   // [o][l stacked r|i]  M-major, padded
  const float* T2   = ws + OFF_T12;
  const float* DD   = ws + OFF_DD;
  const float* g1p  = ws + OFF_G1P;
  const float* beta = ws + OFF_BETA;
  int w0 = blockIdx.x * 64;
  int h  = blockIdx.y;
  int b  = blockIdx.z;
  int t = threadIdx.x, wid = t >> 5, lane = t & 31;
#pragma unroll
  for (int idx = t; idx < 1024; idx += 256) {     // x strip: async 16B/lane
    int i = idx >> 4, w = (idx & 15) << 2;
    async_copy16(&xs[i][w], &x[(((size_t)b * 64 + i) * 256 + h) * 256 + w0 + w]);
  }
  for (int idx = t; idx < 4096; idx += 256) {     // FiLM-scaled spectral rows
    int o = idx >> 6, c = idx & 63;
    int p = c >> 5, l = c & 31;
    a2[o][c] = T2[((size_t)(b * 64 + o)) * 16384 + (size_t)p * 8192 + h * 32 + l]
               * g1p[b * 64 + o];
  }
  wait_async0();
  __syncthreads();
  int m = wid & 3;
  int n0 = (wid >> 2) << 1;
  v8f acc0 = {}, acc1 = {};
#pragma unroll
  for (int k = 0; k < 64; k += 4) {          // bypass: A=bw [o][i], B=xs [i][w]
    v2f a  = ld_mnmaj(bw, 64, m * 16, k);
    v2f b0 = ld_kmaj(&xs[0][0], 64, k, n0 * 16);
    v2f b1 = ld_kmaj(&xs[0][0], 64, k, n0 * 16 + 16);
    acc0 = wmma4(a, b0, acc0);
    acc1 = wmma4(a, b1, acc1);
  }
#pragma unroll
  for (int k = 0; k < 64; k += 4) {          // spectral: A=a2 [o][l], B=DD [l][w]
    v2f a  = ld_mnmaj(&a2[0][0], 68, m * 16, k);
    v2f b0 = ld_kmaj(DD, 256, k, w0 + n0 * 16);
    v2f b1 = ld_kmaj(DD, 256, k, w0 + n0 * 16 + 16);
    acc0 = wmma4(a, b0, acc0);
    acc1 = wmma4(a, b1, acc1);
  }
  int nn = lane & 15, mh = (lane >> 4) * 8;
#pragma unroll
  for (int r = 0; r < 8; ++r) {
    int o = m * 16 + mh + r;
    float add = beta[b * 64 + o] + bb[o];
    size_t rowbase = (((size_t)b * 64 + o) * 256 + h) * 256 + w0;
    out[rowbase + n0 * 16 + nn]      = gelu_tanh(acc0[r] + add);
    out[rowbase + n0 * 16 + 16 + nn] = gelu_tanh(acc1[r] + add);
  }
}

// ---------------------------------------------------------------------------
extern "C" void kernel_launch(void* const* d_in, const int* in_sizes, int n_in,
                              void* d_out, int out_size, void* d_ws, size_t ws_size,
                              hipStream_t stream) {
  (void)in_sizes; (void)n_in; (void)out_size; (void)ws_size;
  const float* x      = (const float*)d_in[0];
  const float* t_emb  = (const float*)d_in[1];
  const float* w_real = (const float*)d_in[2];
  const float* w_imag = (const float*)d_in[3];
  const float* fw1    = (const float*)d_in[4];
  const float* fb1    = (const float*)d_in[5];
  const float* fw2    = (const float*)d_in[6];
  const float* fb2    = (const float*)d_in[7];
  const float* bw     = (const float*)d_in[8];
  const float* bb     = (const float*)d_in[9];
  float* out = (float*)d_out;
  float* ws  = (float*)d_ws;

  k_tables<<<1, 256, 0, stream>>>(ws);
  k_film<<<1, 256, 0, stream>>>(t_emb, fw1, fb1, fw2, fb2, ws);
  k_wt<<<2048, 256, 0, stream>>>(w_real, w_imag, ws);
  k_fwdH<<<dim3(1024, 4), 256, 0, stream>>>(x, ws);
  k_fwdW<<<1024, 256, 0, stream>>>(ws);
  k_mix<<<1024, 256, 0, stream>>>(ws);
  k_invH<<<1024, 256, 0, stream>>>(ws);
  k_final<<<dim3(4, 256, 16), 256, 0, stream>>>(x, bw, bb, ws, out);
}